// HMGNN_5145370821348
// MI455X (gfx1250) — compile-verified
//
#include <hip/hip_runtime.h>

#define NN   50000
#define EE   800000
#define HH   4
#define FF   32
#define EDD  5
#define INN  256
#define NEG_SLOPE 0.2f

typedef __attribute__((ext_vector_type(2))) float v2f;
typedef __attribute__((ext_vector_type(8))) float v8f;

// ---- monotonic f32 <-> u32 encoding for atomic segment-max -----------------
__device__ __forceinline__ unsigned enc_f32(float f) {
  unsigned u = __float_as_uint(f);
  return (u & 0x80000000u) ? ~u : (u | 0x80000000u);
}
__device__ __forceinline__ float dec_f32(unsigned u) {
  return __uint_as_float((u & 0x80000000u) ? (u & 0x7FFFFFFFu) : ~u);
}

// ---- zero-init the atomic accumulators (m_enc, den, packed) ----------------
__global__ void k_zero(float* p, size_t n) {
  size_t i = (size_t)blockIdx.x * blockDim.x + threadIdx.x;
  if (i < n) p[i] = 0.0f;
}

// ---- Kernel 1: feat_src = feat @ W_fc  ([N,256] x [256,128]) via WMMA f32 --
// grid: N/16 blocks, 128 threads (4 waves). Block stages the 16x256 A tile in
// LDS once (row stride padded to 260 floats -> conflict-free banked reads),
// then each wave computes two 16x16 tiles (covering all 128 output columns),
// reusing one A fragment per two WMMAs.
__global__ void k_node_proj(const float* __restrict__ feat,
                            const float* __restrict__ Wfc,
                            float* __restrict__ feat_src) {
  __shared__ float As[16][260];
  const int tid  = threadIdx.x;
  const int wave = tid >> 5;
  const int lane = tid & 31;
  const int row0 = blockIdx.x * 16;

  // cooperative stage of A tile: 16 rows x 256 cols, 8 threads per row
  {
    const int r  = tid >> 3;            // 0..15
    const int c0 = (tid & 7) * 32;      // 0,32,...,224
    const float4* g = (const float4*)(feat + (size_t)(row0 + r) * INN + c0);
    float4* l = (float4*)&As[r][c0];
    #pragma unroll
    for (int q = 0; q < 8; ++q) l[q] = g[q];
  }
  __syncthreads();

  const int m  = lane & 15;
  const int kh = (lane >> 4) * 2;       // 0 or 2 (K sub-offset per lane half)
  const int colA = wave * 32;           // this wave covers cols [colA, colA+32)

  v8f acc0 = {}, acc1 = {};
  #pragma unroll 4
  for (int k = 0; k < INN; k += 4) {
    const int k0 = k + kh;
    v2f a, b0, b1;
    a.x = As[m][k0];
    a.y = As[m][k0 + 1];
    b0.x = Wfc[(size_t)k0 * 128 + colA + m];
    b0.y = Wfc[(size_t)(k0 + 1) * 128 + colA + m];
    b1.x = Wfc[(size_t)k0 * 128 + colA + 16 + m];
    b1.y = Wfc[(size_t)(k0 + 1) * 128 + colA + 16 + m];
    acc0 = __builtin_amdgcn_wmma_f32_16x16x4_f32(false, a, false, b0,
                                                 (short)0, acc0, false, false);
    acc1 = __builtin_amdgcn_wmma_f32_16x16x4_f32(false, a, false, b1,
                                                 (short)0, acc1, false, false);
  }
  const int mbase = (lane < 16) ? 0 : 8;
  const int n = lane & 15;
  #pragma unroll
  for (int r = 0; r < 8; ++r) {
    float* orow = feat_src + (size_t)(row0 + mbase + r) * 128;
    orow[colA + n]      = acc0[r];
    orow[colA + 16 + n] = acc1[r];
  }
}

// ---- Kernel 2: efe = edge_fea @ W_edg + b_edg ; ee = sum(efe*attn_edg) ----
__global__ void k_edge_proj(const float* __restrict__ edge_fea,
                            const float* __restrict__ Wedg,
                            const float* __restrict__ bedg,
                            const float* __restrict__ attn_edg,
                            float* __restrict__ efe,
                            float* __restrict__ ee) {
  const size_t e = (size_t)blockIdx.x * blockDim.x + threadIdx.x;
  if (e >= EE) return;
  float x[EDD];
  #pragma unroll
  for (int i = 0; i < EDD; ++i) x[i] = edge_fea[e * EDD + i];
  #pragma unroll
  for (int h = 0; h < HH; ++h) {
    float s = 0.0f;
    #pragma unroll
    for (int c = 0; c < EDD; ++c) {
      float v = bedg[h * EDD + c];
      #pragma unroll
      for (int i = 0; i < EDD; ++i)
        v = fmaf(x[i], Wedg[i * (HH * EDD) + h * EDD + c], v);
      efe[e * (HH * EDD) + h * EDD + c] = v;
      s = fmaf(v, attn_edg[h * EDD + c], s);
    }
    ee[e * HH + h] = s;
  }
}

// ---- Kernel 3: el / er per (node, head) -----------------------------------
__global__ void k_node_attn(const float* __restrict__ feat_src,
                            const float* __restrict__ attn_l,
                            const float* __restrict__ attn_r,
                            float* __restrict__ el, float* __restrict__ er) {
  const int idx = blockIdx.x * blockDim.x + threadIdx.x;
  if (idx >= NN * HH) return;
  const int n = idx >> 2, h = idx & 3;
  const float* fs = feat_src + (size_t)n * 128 + h * FF;
  float sl = 0.0f, sr = 0.0f;
  #pragma unroll 8
  for (int f = 0; f < FF; ++f) {
    const float v = fs[f];
    sl = fmaf(v, attn_l[h * FF + f], sl);
    sr = fmaf(v, attn_r[h * FF + f], sr);
  }
  el[idx] = sl; er[idx] = sr;
}

// ---- Kernel 4: edge logits + leaky_relu + atomic segment-max --------------
__global__ void k_edge_logit(const int* __restrict__ src, const int* __restrict__ dst,
                             const float* __restrict__ el, const float* __restrict__ er,
                             const float* __restrict__ ee,
                             float* __restrict__ ebuf, unsigned* __restrict__ m_enc) {
  const size_t idx = (size_t)blockIdx.x * blockDim.x + threadIdx.x;
  if (idx >= (size_t)EE * HH) return;
  const int e = (int)(idx >> 2), h = (int)(idx & 3);
  const int s = src[e], d = dst[e];
  float v = el[s * HH + h] + er[d * HH + h] + ee[idx];
  v = (v > 0.0f) ? v : NEG_SLOPE * v;
  ebuf[idx] = v;
  atomicMax(&m_enc[d * HH + h], enc_f32(v));
}

// ---- Kernel 5: ex = exp(e - m[dst]); den[dst] += ex -----------------------
__global__ void k_edge_exp(const int* __restrict__ dst,
                           const unsigned* __restrict__ m_enc,
                           float* __restrict__ ebuf, float* __restrict__ den) {
  const size_t idx = (size_t)blockIdx.x * blockDim.x + threadIdx.x;
  if (idx >= (size_t)EE * HH) return;
  const int e = (int)(idx >> 2), h = (int)(idx & 3);
  const int d = dst[e];
  const float mv = dec_f32(m_enc[d * HH + h]);
  const float ex = __expf(ebuf[idx] - mv);
  ebuf[idx] = ex;
  atomicAdd(&den[d * HH + h], ex);
}

// ---- Kernel 6: a = ex/den[dst]; scatter a*efe and a*feat_src[src] ---------
// Scatters directly into packed [N*H, 40] = concat(ft_f[5], ft[32], pad[3]).
__global__ void k_scatter(const int* __restrict__ src, const int* __restrict__ dst,
                          const float* __restrict__ ebuf, const float* __restrict__ den,
                          const float* __restrict__ efe,
                          const float* __restrict__ feat_src,
                          float* __restrict__ packed) {
  const size_t idx = (size_t)blockIdx.x * blockDim.x + threadIdx.x;
  if (idx >= (size_t)EE * HH) return;
  const int e = (int)(idx >> 2), h = (int)(idx & 3);
  const int d = dst[e], s = src[e];
  const float a = ebuf[idx] / den[d * HH + h];
  float* prow = packed + ((size_t)d * HH + h) * 40;
  const float* erow = efe + (size_t)e * (HH * EDD) + h * EDD;
  #pragma unroll
  for (int c = 0; c < EDD; ++c)
    atomicAdd(&prow[c], a * erow[c]);
  const float4* frow = (const float4*)(feat_src + (size_t)s * 128 + h * FF);
  #pragma unroll
  for (int q = 0; q < FF / 4; ++q) {
    const float4 v = frow[q];
    atomicAdd(&prow[5 + 4 * q + 0], a * v.x);
    atomicAdd(&prow[5 + 4 * q + 1], a * v.y);
    atomicAdd(&prow[5 + 4 * q + 2], a * v.z);
    atomicAdd(&prow[5 + 4 * q + 3], a * v.w);
  }
}

// ---- Kernel 7: rst = packed @ W_out + b_out + bias via WMMA f32 -----------
// M = N*H = 200000, K = 40 (cols 37..39 are zero), Ncols = 32.
// grid: N*H/16, block 64 (2 waves -> two 16-col tiles).
__global__ void k_out_gemm(const float* __restrict__ packed,
                           const float* __restrict__ Wout,
                           const float* __restrict__ b_out,
                           const float* __restrict__ bias,
                           float* __restrict__ out) {
  const int wave = threadIdx.x >> 5;
  const int lane = threadIdx.x & 31;
  const int row0 = blockIdx.x * 16;
  const int col0 = wave * 16;
  const int m  = lane & 15;
  const int kh = (lane >> 4) * 2;

  v8f acc = {};
  const float* arow = packed + (size_t)(row0 + m) * 40;
  #pragma unroll
  for (int k = 0; k < 40; k += 4) {
    const int k0 = k + kh, k1 = k0 + 1;
    v2f a, b;
    a.x = arow[k0];
    a.y = arow[k1];
    b.x = (k0 < EDD + FF) ? Wout[k0 * FF + col0 + m] : 0.0f;
    b.y = (k1 < EDD + FF) ? Wout[k1 * FF + col0 + m] : 0.0f;
    acc = __builtin_amdgcn_wmma_f32_16x16x4_f32(false, a, false, b,
                                                (short)0, acc, false, false);
  }
  const int mbase = (lane < 16) ? 0 : 8;
  const int n = lane & 15;
  #pragma unroll
  for (int r = 0; r < 8; ++r) {
    const size_t row = (size_t)(row0 + mbase + r);   // = node*H + h
    const int h = (int)(row & 3);
    const int col = col0 + n;
    out[row * FF + col] = acc[r] + b_out[col] + bias[h * FF + col];
  }
}

extern "C" void kernel_launch(void* const* d_in, const int* in_sizes, int n_in,
                              void* d_out, int out_size, void* d_ws, size_t ws_size,
                              hipStream_t stream) {
  const float* feat     = (const float*)d_in[0];
  const float* edge_fea = (const float*)d_in[1];
  const int*   src      = (const int*)  d_in[2];
  const int*   dst      = (const int*)  d_in[3];
  const float* W_fc     = (const float*)d_in[4];
  const float* W_edg    = (const float*)d_in[5];
  const float* b_edg    = (const float*)d_in[6];
  const float* attn_l   = (const float*)d_in[7];
  const float* attn_r   = (const float*)d_in[8];
  const float* attn_edg = (const float*)d_in[9];
  const float* W_out    = (const float*)d_in[10];
  const float* b_out    = (const float*)d_in[11];
  const float* bias     = (const float*)d_in[12];
  float* out = (float*)d_out;

  // workspace layout (floats); m/den/packed are adjacent for one zero pass
  float* ws = (float*)d_ws;
  float*    feat_src = ws;                         // N*128      = 6,400,000
  float*    el       = feat_src + (size_t)NN * 128;            // 200,000
  float*    er       = el + (size_t)NN * HH;                   // 200,000
  float*    efe      = er + (size_t)NN * HH;       // E*20       = 16,000,000
  float*    ee       = efe + (size_t)EE * HH * EDD;            // 3,200,000
  float*    ebuf     = ee + (size_t)EE * HH;                   // 3,200,000
  unsigned* m_enc    = (unsigned*)(ebuf + (size_t)EE * HH);    // 200,000
  float*    den      = (float*)(m_enc + (size_t)NN * HH);      // 200,000
  float*    packed   = den + (size_t)NN * HH;      // N*H*40    = 8,000,000

  // zero m_enc + den + packed (contiguous, 8.4M elements)
  const size_t zlen = (size_t)NN * HH * 2 + (size_t)NN * HH * 40;
  k_zero<<<(unsigned)((zlen + 255) / 256), 256, 0, stream>>>((float*)m_enc, zlen);

  // node projection GEMM (WMMA f32 16x16x4, LDS-staged A tile)
  k_node_proj<<<NN / 16, 128, 0, stream>>>(feat, W_fc, feat_src);

  // edge projection + ee
  k_edge_proj<<<(EE + 255) / 256, 256, 0, stream>>>(edge_fea, W_edg, b_edg,
                                                    attn_edg, efe, ee);
  // node attention logits
  k_node_attn<<<(NN * HH + 255) / 256, 256, 0, stream>>>(feat_src, attn_l, attn_r, el, er);

  // edge logits + segment-max
  const size_t eh = (size_t)EE * HH;
  k_edge_logit<<<(unsigned)((eh + 255) / 256), 256, 0, stream>>>(src, dst, el, er, ee,
                                                                 ebuf, m_enc);
  // exp + denominator
  k_edge_exp<<<(unsigned)((eh + 255) / 256), 256, 0, stream>>>(dst, m_enc, ebuf, den);

  // weighted scatter-aggregation into packed [N*H, 40]
  k_scatter<<<(unsigned)((eh + 255) / 256), 256, 0, stream>>>(src, dst, ebuf, den,
                                                              efe, feat_src, packed);
  // output GEMM (WMMA f32 16x16x4) with bias epilogue
  k_out_gemm<<<(NN * HH) / 16, 64, 0, stream>>>(packed, W_out, b_out, bias, out);
}